// GCN_36593121362326
// MI455X (gfx1250) — compile-verified
//
#include <hip/hip_runtime.h>
#include <hip/hip_bf16.h>
#include <math.h>

// ---------------------------------------------------------------------------
// GCN forward on MI455X (gfx1250).
//   h1 = relu(GCNagg(X @ W1) + b1)
//   h2 = relu(GCNagg(h1 @ W2) + b2)
//   out = sigmoid(h2 @ Wl + bl)
// GEMMs use V_WMMA_F32_16X16X4_F32 (fp32 precision on the matrix pipe).
// Edge aggregation: wave-per-edge (32 lanes x 2 channels) coalesced gather
// + global_atomic_add_f32 into a 12.8MB accumulator that lives in L2.
// ---------------------------------------------------------------------------

#define N_NODES 50000
#define N_EDGES 1600000
#define IN_DIM  128
#define HID     64

typedef __attribute__((ext_vector_type(2))) float v2f;
typedef __attribute__((ext_vector_type(8))) float v8f;

// ---------------- degree / normalization ----------------

__global__ void k_init_deg(float* __restrict__ deg, int n) {
    int i = blockIdx.x * blockDim.x + threadIdx.x;
    if (i < n) deg[i] = 1.0f;   // self-loop weight
}

__global__ void k_deg_scatter(const int* __restrict__ dst,
                              const float* __restrict__ w,
                              float* __restrict__ deg, int E) {
    int i = blockIdx.x * blockDim.x + threadIdx.x;
    if (i < E) atomicAdd(&deg[dst[i]], w[i]);
}

__global__ void k_make_dis(const float* __restrict__ deg,
                           float* __restrict__ dis, int n) {
    int i = blockIdx.x * blockDim.x + threadIdx.x;
    if (i < n) {
        float d = deg[i];
        dis[i] = (d > 0.0f) ? rsqrtf(d) : 0.0f;
    }
}

__global__ void k_zero(float* __restrict__ p, long n) {
    long i = (long)blockIdx.x * blockDim.x + threadIdx.x;
    if (i < n) p[i] = 0.0f;
}

// ---------------- fp32 WMMA GEMM: C[n,64] = A[n,K] @ W[K,64] ----------------
// blockDim = 128 (4 waves). Wave w computes the 16x16 tile at
// rows [16*blockIdx.x, +16), cols [16*w, +16). K stepped by 4 via
// v_wmma_f32_16x16x4_f32.

__global__ void __launch_bounds__(128)
k_gemm_wmma_n64(const float* __restrict__ A, const float* __restrict__ W,
                float* __restrict__ C, int n, int K) {
    const int lane  = threadIdx.x & 31;
    const int wave  = threadIdx.x >> 5;          // 0..3 -> N tile
    const int rbase = blockIdx.x * 16;
    const int cbase = wave * 16;
    const int m     = lane & 15;                 // row within tile (A), col (B)
    const int khalf = (lane >> 4) ? 2 : 0;       // hi half-wave: K+2,K+3

    // clamp row so OOB lanes read valid memory (n=50000 is 16-aligned anyway)
    int arow = rbase + m; if (arow >= n) arow = n - 1;
    const float* arowp = A + (size_t)arow * K;

    v8f acc = {};
    for (int k = 0; k < K; k += 4) {
        // A fragment: A[arow][k+khalf], A[arow][k+khalf+1]  (contiguous pair)
        const float* ap = arowp + k + khalf;
        v2f a; a.x = ap[0]; a.y = ap[1];
        // B fragment: W[k+khalf][cbase+m], W[k+khalf+1][cbase+m]
        const float* bp = W + (size_t)(k + khalf) * HID + cbase + m;
        v2f b; b.x = bp[0]; b.y = bp[HID];
        acc = __builtin_amdgcn_wmma_f32_16x16x4_f32(
                  false, a, false, b, (short)0, acc, false, false);
    }

    // C/D layout: VGPR r -> row rbase + (lane>=16 ? 8 : 0) + r, col cbase+(lane&15)
    const int row0 = rbase + ((lane >> 4) ? 8 : 0);
    const int col  = cbase + (lane & 15);
    float* cp = C + (size_t)row0 * HID + col;
#pragma unroll
    for (int r = 0; r < 8; ++r) {
        if (row0 + r < n) cp[(size_t)r * HID] = acc[r];
    }
}

// ---------------- edge scatter: agg[dst] += h[src] * norm ----------------
// One wave (32 lanes) per edge; each lane owns 2 contiguous channels.

__global__ void k_edge_agg(const int* __restrict__ src,
                           const int* __restrict__ dst,
                           const float* __restrict__ w,
                           const float* __restrict__ dis,
                           const float* __restrict__ h,
                           float* __restrict__ agg, int E) {
    long t = (long)blockIdx.x * blockDim.x + threadIdx.x;
    int  e = (int)(t >> 5);
    int  lane = (int)(t & 31);
    if (e >= E) return;
    int s = src[e];
    int d = dst[e];
    float norm = dis[s] * w[e] * dis[d];
    const float2 hv = *(const float2*)(h + (size_t)s * HID + lane * 2);
    float* ap = agg + (size_t)d * HID + lane * 2;
    atomicAdd(ap + 0, hv.x * norm);
    atomicAdd(ap + 1, hv.y * norm);
}

// ---------------- self-loop + bias + ReLU ----------------
// agg[i] = relu(agg[i] + h[i]*dis[i]^2 + b)

__global__ void k_self_bias_relu(const float* __restrict__ h,
                                 const float* __restrict__ dis,
                                 const float* __restrict__ b,
                                 float* __restrict__ agg, int n) {
    long t = (long)blockIdx.x * blockDim.x + threadIdx.x;
    int  i = (int)(t >> 5);
    int  lane = (int)(t & 31);
    if (i >= n) return;
    float di = dis[i];
    float sc = di * di;                 // self-loop norm (weight 1)
    size_t base = (size_t)i * HID + lane * 2;
    float2 hv = *(const float2*)(h + base);
    float2 av = *(const float2*)(agg + base);
    float x0 = av.x + hv.x * sc + b[lane * 2 + 0];
    float x1 = av.y + hv.y * sc + b[lane * 2 + 1];
    float2 o; o.x = x0 > 0.0f ? x0 : 0.0f; o.y = x1 > 0.0f ? x1 : 0.0f;
    *(float2*)(agg + base) = o;
}

// ---------------- head: out[i] = sigmoid(dot(x[i], Wl) + bl) ----------------
// One wave per node, wave32 xor-shuffle reduction.

__global__ void k_head_sigmoid(const float* __restrict__ x,
                               const float* __restrict__ Wl,
                               const float* __restrict__ bl,
                               float* __restrict__ out, int n) {
    long t = (long)blockIdx.x * blockDim.x + threadIdx.x;
    int  i = (int)(t >> 5);
    int  lane = (int)(t & 31);
    if (i >= n) return;
    const float* xp = x + (size_t)i * HID;
    float acc = xp[lane] * Wl[lane] + xp[lane + 32] * Wl[lane + 32];
#pragma unroll
    for (int off = 16; off > 0; off >>= 1)
        acc += __shfl_xor(acc, off, 32);
    if (lane == 0) {
        float z = acc + bl[0];
        out[i] = 1.0f / (1.0f + expf(-z));
    }
}

// ---------------------------------------------------------------------------

extern "C" void kernel_launch(void* const* d_in, const int* in_sizes, int n_in,
                              void* d_out, int out_size, void* d_ws, size_t ws_size,
                              hipStream_t stream) {
    const float* X  = (const float*)d_in[0];
    const int*   ei = (const int*)d_in[1];      // [2, E] int32
    const float* w  = (const float*)d_in[2];
    const float* W1 = (const float*)d_in[3];
    const float* b1 = (const float*)d_in[4];
    const float* W2 = (const float*)d_in[5];
    const float* b2 = (const float*)d_in[6];
    const float* Wl = (const float*)d_in[7];
    const float* bl = (const float*)d_in[8];
    float* out = (float*)d_out;

    const int n = N_NODES;
    const int E = in_sizes[2];                  // edge count from weight array
    const int* src = ei;
    const int* dst = ei + E;

    // workspace carve-up (all fp32): deg[n] dis[n] h[n*64] agg[n*64] ~ 26 MB
    float* deg = (float*)d_ws;
    float* dis = deg + n;
    float* h   = dis + n;
    float* agg = h + (size_t)n * HID;

    const int B = 256;
    const long nodeCh = (long)n * 32;           // (node, lane) threads
    const long edgeCh = (long)E * 32;           // (edge, lane) threads

    // normalization (shared by both layers)
    k_init_deg   <<<(n + B - 1) / B, B, 0, stream>>>(deg, n);
    k_deg_scatter<<<(E + B - 1) / B, B, 0, stream>>>(dst, w, deg, E);
    k_make_dis   <<<(n + B - 1) / B, B, 0, stream>>>(deg, dis, n);

    // ---- layer 1 ----
    k_gemm_wmma_n64<<<(n + 15) / 16, 128, 0, stream>>>(X, W1, h, n, IN_DIM);
    k_zero<<<(long)((long)n * HID + B - 1) / B, B, 0, stream>>>(agg, (long)n * HID);
    k_edge_agg<<<(edgeCh + B - 1) / B, B, 0, stream>>>(src, dst, w, dis, h, agg, E);
    k_self_bias_relu<<<(nodeCh + B - 1) / B, B, 0, stream>>>(h, dis, b1, agg, n);

    // ---- layer 2 ----  (agg holds x1; gemm reads it before agg is re-zeroed)
    k_gemm_wmma_n64<<<(n + 15) / 16, 128, 0, stream>>>(agg, W2, h, n, HID);
    k_zero<<<(long)((long)n * HID + B - 1) / B, B, 0, stream>>>(agg, (long)n * HID);
    k_edge_agg<<<(edgeCh + B - 1) / B, B, 0, stream>>>(src, dst, w, dis, h, agg, E);
    k_self_bias_relu<<<(nodeCh + B - 1) / B, B, 0, stream>>>(h, dis, b2, agg, n);

    // ---- head ----
    k_head_sigmoid<<<(nodeCh + B - 1) / B, B, 0, stream>>>(agg, Wl, bl, out, n);
}